// ODEG_8942121911067
// MI455X (gfx1250) — compile-verified
//
#include <hip/hip_runtime.h>
#include <hip/hip_bf16.h>

// ---------------------------------------------------------------------------
// Shapes (from reference): B=32, N=512, T=24, F=64, Fa=74
//   x   : [32,512,24,64]  fp32
//   adj : [512,512]       fp32
//   out : [32,512,24,74]  fp32
// Math: out = relu(0.5*x_aug + 0.125*sig(alpha)*xa + 0.25*Sx*rowsumW + 0.25*xw2)
//   xa = adj @ x (over nodes) -> fp32 WMMA GEMM, dominant cost (~30 GFLOP).
// ---------------------------------------------------------------------------

#define Bsz 32
#define Nn  512
#define Tt  24
#define Ff  64
#define Fa  74
#define TF  (Tt * Ff)          // 1536

typedef float v2f __attribute__((ext_vector_type(2)));
typedef float v8f __attribute__((ext_vector_type(8)));

// ws layout (floats): [0,512) sigAlpha | [512,592) rowsumW (74 used) |
//                     [592,1168) W2 (24x24) | [1280, 1280+B*N*TF) R
#define WS_SA   0
#define WS_RSW  512
#define WS_W2   592
#define WS_R    1280

// ---------------------------------------------------------------------------
// Kernel 1: tiny prep — sigmoid(alpha), rowsumW, W2.  One block of 256.
// ---------------------------------------------------------------------------
__global__ void ode_prep(const float* __restrict__ alpha,
                         const float* __restrict__ w,
                         const float* __restrict__ d,
                         const float* __restrict__ w2,
                         const float* __restrict__ d2,
                         float* __restrict__ ws) {
    __shared__ float cs[Fa];
    const int tid = threadIdx.x;

    for (int e = tid; e < Nn; e += 256)
        ws[WS_SA + e] = 1.0f / (1.0f + __expf(-alpha[e]));

    if (tid < Fa) {                       // colsum_w[p] = sum_n w[n,p]
        float s = 0.f;
        for (int n = 0; n < Fa; ++n) s += w[n * Fa + tid];
        cs[tid] = s;
    }
    __syncthreads();
    if (tid < Fa) {                       // rowsumW[m] = sum_p w[m,p]*dc[p]*cs[p]
        float s = 0.f;
        for (int p = 0; p < Fa; ++p) {
            float dc = fminf(fmaxf(d[p], 0.f), 1.f);
            s += w[tid * Fa + p] * dc * cs[p];
        }
        ws[WS_RSW + tid] = s;
    }
    for (int e = tid; e < Tt * Tt; e += 256) {   // W2[k,n]=sum_p w2[k,p]*d2c[p]*w2[n,p]
        int k = e / Tt, n = e % Tt;
        float s = 0.f;
        for (int p = 0; p < Tt; ++p) {
            float dc = fminf(fmaxf(d2[p], 0.f), 1.f);
            s += w2[k * Tt + p] * dc * w2[n * Tt + p];
        }
        ws[WS_W2 + e] = s;
    }
}

// ---------------------------------------------------------------------------
// Kernel 2: per-(b,i) temporal mixing + row sums.
//   R = 0.5*x + 0.25*xw2 + 0.25*Sx[t]*rowsumW[f]   (features 0..63, to ws)
//   out[...,64..73] = relu(0.25*Sx[t]*rowsumW[f])  (final, zero-feature tail)
// ---------------------------------------------------------------------------
__global__ void ode_temporal(const float* __restrict__ x,
                             float* __restrict__ ws,
                             float* __restrict__ out) {
    __shared__ float xl[TF];
    __shared__ float W2l[Tt * Tt];
    __shared__ float Sxl[Tt];

    const int tid = threadIdx.x;
    const int bi  = blockIdx.x;               // b*512 + i
    const size_t base = (size_t)bi * TF;

    for (int p = 0; p < 6; ++p) xl[tid + p * 256] = x[base + tid + p * 256];
    for (int e = tid; e < Tt * Tt; e += 256) W2l[e] = ws[WS_W2 + e];
    __syncthreads();

    if (tid < Tt) {                           // Sx[t] = sum_f x[t,f]
        float s = 0.f;
        const float* row = &xl[tid * Ff];
        for (int f = 0; f < Ff; ++f) s += row[f];
        Sxl[tid] = s;
    }
    __syncthreads();

    for (int p = 0; p < 6; ++p) {
        const int e = tid + p * 256;
        const int t = e >> 6, f = e & 63;
        float s = 0.f;                        // xw2 = sum_k x[k,f]*W2[k,t]
        #pragma unroll
        for (int k = 0; k < Tt; ++k) s += xl[k * Ff + f] * W2l[k * Tt + t];
        float r = 0.5f * xl[e] + 0.25f * s + 0.25f * Sxl[t] * ws[WS_RSW + f];
        ws[WS_R + base + e] = r;
    }

    if (tid < Tt * (Fa - Ff)) {               // 240 tail outputs
        const int t = tid / (Fa - Ff), fz = Ff + tid % (Fa - Ff);
        const float v = 0.25f * Sxl[t] * ws[WS_RSW + fz];
        out[((size_t)bi * Tt + t) * Fa + fz] = fmaxf(v, 0.f);
    }
}

// ---------------------------------------------------------------------------
// Kernel 3: adjacency GEMM on WMMA fp32 pipes + fused epilogue.
//   xa[b] = adj @ x[b]  :  [512x512] @ [512x1536], one batch per blockIdx.z.
//   Block tile: M=64, N=256, K-tile=32. 8 wave32 waves; wave w owns
//   2 M-subtiles ((w&1)*32 + {0,16}) x 4 N-subtiles ((w>>1)*64 + j*16):
//   8 WMMAs per k-step from 6 fragment loads (amortized LDS reads).
//   Double-buffered LDS staging + global_prefetch two tiles ahead.
//   Epilogue: out = relu(0.125*sigAlpha[i]*xa + R).
// ---------------------------------------------------------------------------
#define MT  64
#define NTc 256
#define KT  32
#define LDP 33          // stride-33 rows: conflict-free LDS access

__global__ void __launch_bounds__(256)
ode_gemm_wmma(const float* __restrict__ adj,
              const float* __restrict__ x,
              const float* __restrict__ ws,
              float* __restrict__ out) {
    __shared__ float A_lds[2 * MT * LDP];    // A[m][k] row-major, stride 33
    __shared__ float B_lds[2 * NTc * LDP];   // B transposed: [n][k], stride 33

    const int tid   = threadIdx.x;
    const int lane  = tid & 31;
    const int wave  = tid >> 5;
    const int l15   = lane & 15;
    const int lhalf = lane >> 4;

    const int mgrp  = (wave & 1) * 32;        // wave's M group (2 subtiles of 16)
    const int nbase = (wave >> 1) * 64;       // wave's N strip (4 subtiles of 16)

    const int blockM = blockIdx.x * MT;
    const int ncol0  = blockIdx.y * NTc;
    const int b      = blockIdx.z;
    const size_t xb  = (size_t)b * Nn * TF;

    // staging-thread coordinates
    const int kqA = (tid & 7) * 4;            // A: col group within K-tile
    const int m0  = tid >> 3;                 // A: row 0..31 (2 passes)
    const int cqB = (tid & 63) * 4;           // B: col group within N-tile
    const int kB  = tid >> 6;                 // B: row 0..3 (8 passes)

    v8f acc[2][4];
    #pragma unroll
    for (int mi = 0; mi < 2; ++mi)
        #pragma unroll
        for (int j = 0; j < 4; ++j) acc[mi][j] = v8f{0};

    float4 aS[2], bS[8];

    // ---- stage tile 0 ----
    #pragma unroll
    for (int p = 0; p < 2; ++p)
        aS[p] = *(const float4*)&adj[(size_t)(blockM + p * 32 + m0) * Nn + kqA];
    #pragma unroll
    for (int p = 0; p < 8; ++p)
        bS[p] = *(const float4*)&x[xb + (size_t)(p * 4 + kB) * TF + ncol0 + cqB];
    #pragma unroll
    for (int p = 0; p < 2; ++p) {
        float* dst = &A_lds[(p * 32 + m0) * LDP + kqA];
        dst[0] = aS[p].x; dst[1] = aS[p].y; dst[2] = aS[p].z; dst[3] = aS[p].w;
    }
    #pragma unroll
    for (int p = 0; p < 8; ++p) {
        const int k = p * 4 + kB;
        B_lds[(cqB + 0) * LDP + k] = bS[p].x;
        B_lds[(cqB + 1) * LDP + k] = bS[p].y;
        B_lds[(cqB + 2) * LDP + k] = bS[p].z;
        B_lds[(cqB + 3) * LDP + k] = bS[p].w;
    }
    __syncthreads();

    int cur = 0;
    for (int it = 0; it < Nn / KT; ++it) {
        const int k0n = (it + 1) * KT;
        // ---- issue global loads for next tile (overlap with WMMA below) ----
        if (it < Nn / KT - 1) {
            #pragma unroll
            for (int p = 0; p < 2; ++p)
                aS[p] = *(const float4*)&adj[(size_t)(blockM + p * 32 + m0) * Nn + k0n + kqA];
            #pragma unroll
            for (int p = 0; p < 8; ++p)
                bS[p] = *(const float4*)&x[xb + (size_t)(k0n + p * 4 + kB) * TF + ncol0 + cqB];
            if (it < Nn / KT - 2) {           // global_prefetch_b8 two tiles ahead
                __builtin_prefetch(&adj[(size_t)(blockM + m0) * Nn + k0n + KT + kqA], 0, 0);
                __builtin_prefetch(&x[xb + (size_t)(k0n + KT + kB) * TF + ncol0 + cqB], 0, 0);
            }
        }

        // ---- compute: 8 k-steps x 8 WMMAs from current buffer ----
        const float* Ac = A_lds + cur * (MT * LDP);
        const float* Bc = B_lds + cur * (NTc * LDP);
        #pragma unroll
        for (int kk = 0; kk < KT; kk += 4) {
            const int kb = kk + 2 * lhalf;    // lanes 0-15: K=kk,kk+1; 16-31: +2,+3
            v2f a[2];
            #pragma unroll
            for (int mi = 0; mi < 2; ++mi) {
                const float* ap = &Ac[(mgrp + mi * 16 + l15) * LDP + kb];
                a[mi].x = ap[0]; a[mi].y = ap[1];
            }
            #pragma unroll
            for (int j = 0; j < 4; ++j) {
                const float* bp = &Bc[(nbase + j * 16 + l15) * LDP + kb];
                v2f bf; bf.x = bp[0]; bf.y = bp[1];
                acc[0][j] = __builtin_amdgcn_wmma_f32_16x16x4_f32(
                    false, a[0], false, bf, (short)0, acc[0][j], false, false);
                acc[1][j] = __builtin_amdgcn_wmma_f32_16x16x4_f32(
                    false, a[1], false, bf, (short)0, acc[1][j], false, false);
            }
        }

        // ---- store staged registers into the other buffer ----
        if (it < Nn / KT - 1) {
            float* An = A_lds + (cur ^ 1) * (MT * LDP);
            float* Bn = B_lds + (cur ^ 1) * (NTc * LDP);
            #pragma unroll
            for (int p = 0; p < 2; ++p) {
                float* dst = &An[(p * 32 + m0) * LDP + kqA];
                dst[0] = aS[p].x; dst[1] = aS[p].y; dst[2] = aS[p].z; dst[3] = aS[p].w;
            }
            #pragma unroll
            for (int p = 0; p < 8; ++p) {
                const int k = p * 4 + kB;
                Bn[(cqB + 0) * LDP + k] = bS[p].x;
                Bn[(cqB + 1) * LDP + k] = bS[p].y;
                Bn[(cqB + 2) * LDP + k] = bS[p].z;
                Bn[(cqB + 3) * LDP + k] = bS[p].w;
            }
        }
        __syncthreads();
        cur ^= 1;
    }

    // ---- epilogue: out = relu(0.125*sigAlpha[i]*xa + R) ----
    #pragma unroll
    for (int mi = 0; mi < 2; ++mi) {
        #pragma unroll
        for (int r = 0; r < 8; ++r) {
            const int i = blockM + mgrp + mi * 16 + 8 * lhalf + r; // C: m = r + 8*hi
            const float sav = 0.125f * ws[WS_SA + i];
            const size_t rowR = (size_t)(b * Nn + i) * TF;
            const size_t rowO = (size_t)(b * Nn + i) * Tt;
            #pragma unroll
            for (int j = 0; j < 4; ++j) {
                const int c = ncol0 + nbase + j * 16 + l15;        // 0..1535
                const int t = c >> 6, f = c & 63;
                const float v = sav * acc[mi][j][r] + ws[WS_R + rowR + c];
                out[(rowO + t) * Fa + f] = fmaxf(v, 0.f);
            }
        }
    }
}

// ---------------------------------------------------------------------------
extern "C" void kernel_launch(void* const* d_in, const int* in_sizes, int n_in,
                              void* d_out, int out_size, void* d_ws, size_t ws_size,
                              hipStream_t stream) {
    const float* x     = (const float*)d_in[0];
    const float* adj   = (const float*)d_in[1];
    const float* alpha = (const float*)d_in[2];
    const float* w     = (const float*)d_in[3];
    const float* d     = (const float*)d_in[4];
    const float* w2    = (const float*)d_in[5];
    const float* d2    = (const float*)d_in[6];
    float* out = (float*)d_out;
    float* ws  = (float*)d_ws;   // needs 1280 + 32*512*1536 floats (~101 MB)

    ode_prep<<<1, 256, 0, stream>>>(alpha, w, d, w2, d2, ws);
    ode_temporal<<<Bsz * Nn, 256, 0, stream>>>(x, ws, out);
    dim3 grid(Nn / MT, TF / NTc, Bsz);       // (8, 6, 32)
    ode_gemm_wmma<<<grid, 256, 0, stream>>>(adj, x, ws, out);
}